// DGCN_20409684591159
// MI455X (gfx1250) — compile-verified
//
#include <hip/hip_runtime.h>

typedef float v2f __attribute__((ext_vector_type(2)));
typedef float v8f __attribute__((ext_vector_type(8)));

#define N_NODES 50000
#define N_EDGES 800000
#define DIM     128
#define MTILES  (N_NODES / 16)   // 3125 exactly
#define LDSTRIDE 132             // padded row stride (floats) to avoid LDS bank conflicts

// ---------------- elementwise / graph kernels ----------------

__global__ void k_init_deg(float* __restrict__ deg) {
    int i = blockIdx.x * blockDim.x + threadIdx.x;
    if (i < N_NODES) deg[i] = 1.0f;   // self-loop
}

__global__ void k_count_deg(const int* __restrict__ dst, float* __restrict__ deg) {
    int e = blockIdx.x * blockDim.x + threadIdx.x;
    if (e < N_EDGES) atomicAdd(&deg[dst[e]], 1.0f);
}

__global__ void k_dinv(const float* __restrict__ deg, float* __restrict__ dinv) {
    int i = blockIdx.x * blockDim.x + threadIdx.x;
    if (i < N_NODES) dinv[i] = rsqrtf(deg[i]);
}

__global__ void k_zero4(float4* __restrict__ p, int n4) {
    int i = blockIdx.x * blockDim.x + threadIdx.x;
    if (i < n4) p[i] = make_float4(0.f, 0.f, 0.f, 0.f);
}

// Wcat[0:128]   = Wt + I
// Wcat[128:256] = W0 - Wt
__global__ void k_build_wcat(const float* __restrict__ W0, const float* __restrict__ Wt,
                             float* __restrict__ Wcat) {
    int idx = blockIdx.x * blockDim.x + threadIdx.x;   // 0 .. 256*128-1
    if (idx >= 256 * DIM) return;
    int k = idx / DIM, n = idx % DIM;
    if (k < DIM) {
        Wcat[idx] = Wt[k * DIM + n] + ((k == n) ? 1.0f : 0.0f);
    } else {
        int kk = k - DIM;
        Wcat[idx] = W0[kk * DIM + n] - Wt[kk * DIM + n];
    }
}

// one wave per edge: agg[dst] += xw[src] * dinv[src]*dinv[dst]
__global__ void k_scatter(const int* __restrict__ src, const int* __restrict__ dst,
                          const float* __restrict__ dinv, const float* __restrict__ xw,
                          float* __restrict__ agg) {
    int e = blockIdx.x * (blockDim.x >> 5) + (threadIdx.x >> 5);
    if (e >= N_EDGES) return;
    int lane = threadIdx.x & 31;
    int s = src[e], d = dst[e];
    float w = dinv[s] * dinv[d];
    float4 v = *(const float4*)(xw + (size_t)s * DIM + lane * 4);
    float* p = agg + (size_t)d * DIM + lane * 4;
    atomicAdd(p + 0, v.x * w);
    atomicAdd(p + 1, v.y * w);
    atomicAdd(p + 2, v.z * w);
    atomicAdd(p + 3, v.w * w);
}

// h_neigh = agg + xw * dinv^2 (self-loop msg) + bc   (in place on agg)
__global__ void k_selfloop_bias(float4* __restrict__ agg, const float4* __restrict__ xw,
                                const float* __restrict__ dinv, const float4* __restrict__ bc4) {
    int idx = blockIdx.x * blockDim.x + threadIdx.x;   // over N*D/4 float4s
    if (idx >= N_NODES * (DIM / 4)) return;
    int row = idx >> 5;           // 32 float4 per row
    int c4  = idx & 31;
    float di = dinv[row];
    float w = di * di;
    float4 a = agg[idx], v = xw[idx], b = bc4[c4];
    a.x += v.x * w + b.x;
    a.y += v.y * w + b.y;
    a.z += v.z * w + b.z;
    a.w += v.w * w + b.w;
    agg[idx] = a;
}

// ---------------- WMMA GEMM ----------------
// out[N,128] = A @ Wg  where A is [A0 | A1] along K (A1 used iff ktot==256).
// One wave computes a 16x128 output strip. Weights staged in LDS (padded).
__global__ void k_gemm_wmma(const float* __restrict__ A0, const float* __restrict__ A1,
                            const float* __restrict__ Wg, float* __restrict__ out,
                            int ktot) {
    extern __shared__ float lds[];   // ktot * LDSTRIDE floats

    // cooperative stage of weights: 128 threads, float4 granularity
    int nf4 = ktot * (DIM / 4);      // # of float4 in Wg
    for (int i = threadIdx.x; i < nf4; i += blockDim.x) {
        int k  = i >> 5;             // 32 float4 per source row
        int n4 = i & 31;
        float4 v = *(const float4*)(Wg + (size_t)k * DIM + n4 * 4);
        *(float4*)(lds + (size_t)k * LDSTRIDE + n4 * 4) = v;   // 132*4B row stride, 16B aligned
    }
    __syncthreads();

    int mtile = blockIdx.x * 4 + (threadIdx.x >> 5);
    if (mtile >= MTILES) return;

    int lane = threadIdx.x & 31;
    int hi   = lane >> 4;            // 0: lanes 0-15, 1: lanes 16-31
    int mrow = mtile * 16 + (lane & 15);
    int koff = hi * 2;               // K-pair offset within 4-wide K step
    int ncol = lane & 15;

    v8f acc[8];
#pragma unroll
    for (int t = 0; t < 8; ++t) acc[t] = (v8f)(0.0f);

    for (int k = 0; k < ktot; k += 4) {
        const float* Asrc = (k < DIM) ? A0 : A1;
        int ka = (k < DIM) ? k : (k - DIM);
        v2f a = *(const v2f*)(Asrc + (size_t)mrow * DIM + ka + koff);
        const float* brow0 = lds + (size_t)(k + koff) * LDSTRIDE;
        const float* brow1 = brow0 + LDSTRIDE;
#pragma unroll
        for (int t = 0; t < 8; ++t) {
            v2f b;
            b.x = brow0[t * 16 + ncol];
            b.y = brow1[t * 16 + ncol];
            acc[t] = __builtin_amdgcn_wmma_f32_16x16x4_f32(
                false, a, false, b, (short)0, acc[t], false, false);
        }
    }

    // C/D layout: VGPR j -> row (j + 8*hi), col = lane&15 within each 16-wide tile
    int mbase = mtile * 16 + hi * 8;
#pragma unroll
    for (int t = 0; t < 8; ++t) {
#pragma unroll
        for (int j = 0; j < 8; ++j) {
            out[(size_t)(mbase + j) * DIM + t * 16 + ncol] = acc[t][j];
        }
    }
}

// ---------------- launcher ----------------

extern "C" void kernel_launch(void* const* d_in, const int* in_sizes, int n_in,
                              void* d_out, int out_size, void* d_ws, size_t ws_size,
                              hipStream_t stream) {
    (void)in_sizes; (void)n_in; (void)out_size; (void)ws_size;

    const float* x   = (const float*)d_in[0];
    const int*   ei  = (const int*)d_in[1];       // [2, E]: row0 = src, row1 = dst
    const float* Wc  = (const float*)d_in[2];
    const float* bc  = (const float*)d_in[3];
    const float* W0  = (const float*)d_in[4];
    const float* Wt  = (const float*)d_in[5];
    float* outp      = (float*)d_out;

    const int* src = ei;
    const int* dst = ei + N_EDGES;

    // workspace layout (256B aligned)
    char* ws = (char*)d_ws;
    size_t off = 0;
    auto take = [&](size_t bytes) { char* p = ws + off; off = (off + bytes + 255) & ~(size_t)255; return p; };
    float* deg  = (float*)take((size_t)N_NODES * 4);
    float* dinv = (float*)take((size_t)N_NODES * 4);
    float* xw   = (float*)take((size_t)N_NODES * DIM * 4);
    float* agg  = (float*)take((size_t)N_NODES * DIM * 4);
    float* Wcat = (float*)take((size_t)256 * DIM * 4);

    // 1) degrees (self-loop init) + dinv
    k_init_deg<<<(N_NODES + 255) / 256, 256, 0, stream>>>(deg);
    k_count_deg<<<(N_EDGES + 255) / 256, 256, 0, stream>>>(dst, deg);
    k_dinv<<<(N_NODES + 255) / 256, 256, 0, stream>>>(deg, dinv);

    // 2) zero agg
    int n4 = N_NODES * (DIM / 4);
    k_zero4<<<(n4 + 255) / 256, 256, 0, stream>>>((float4*)agg, n4);

    // 3) GEMM1: xw = x @ Wc  (K = 128)
    int gemm_blocks = (MTILES + 3) / 4;
    k_gemm_wmma<<<gemm_blocks, 128, 128 * LDSTRIDE * 4, stream>>>(x, x, Wc, xw, 128);

    // 4) edge scatter (1 wave per edge, 8 waves per block)
    k_scatter<<<(N_EDGES + 7) / 8, 256, 0, stream>>>(src, dst, dinv, xw, agg);

    // 5) h_neigh = agg + xw*dinv^2 + bc  (in place)
    k_selfloop_bias<<<(n4 + 255) / 256, 256, 0, stream>>>(
        (float4*)agg, (const float4*)xw, dinv, (const float4*)bc);

    // 6) Wcat = [[Wt + I], [W0 - Wt]]
    k_build_wcat<<<(256 * DIM + 255) / 256, 256, 0, stream>>>(W0, Wt, Wcat);

    // 7) out = [h_neigh | x] @ Wcat  (K = 256)
    k_gemm_wmma<<<gemm_blocks, 128, 256 * LDSTRIDE * 4, stream>>>(agg, x, Wcat, outp, 256);
}